// BipartiteSAGEConv_1099511628127
// MI455X (gfx1250) — compile-verified
//
#include <hip/hip_runtime.h>

typedef __attribute__((ext_vector_type(2))) float v2f;
typedef __attribute__((ext_vector_type(8))) float v8f;

#define DIM   128
#define OUTD  128
#define LDST  130   // LDS row stride (pad: breaks 16-way bank conflicts, keeps 8B align)

// ---------------------------------------------------------------------------
// Kernel 0: zero the scatter workspace (summed[num_dst*128] ++ count[num_dst])
// ---------------------------------------------------------------------------
__global__ void zero_ws_kernel(float* __restrict__ p, long n) {
  long i      = (long)blockIdx.x * blockDim.x + threadIdx.x;
  long stride = (long)gridDim.x * blockDim.x;
  long n4     = n >> 2;
  float4* p4  = (float4*)p;
  for (long j = i; j < n4; j += stride) p4[j] = make_float4(0.f, 0.f, 0.f, 0.f);
  for (long j = (n4 << 2) + i; j < n; j += stride) p[j] = 0.f;
}

// ---------------------------------------------------------------------------
// Kernel 1: edge gather + scatter-sum (+ degree count). One wave32 per edge:
// lane*4 float4 covers the full 128-float row with one coalesced 512B read.
// unsafeAtomicAdd -> global_atomic_add_f32 (L2-resident destination).
// ---------------------------------------------------------------------------
__device__ __forceinline__ void fatomic_add(float* p, float v) {
  unsafeAtomicAdd(p, v);
}

__global__ void scatter_mean_accum(const float* __restrict__ x_src,
                                   const int*   __restrict__ esrc,
                                   const int*   __restrict__ edst,
                                   float*       __restrict__ summed,
                                   float*       __restrict__ cnt,
                                   int E) {
  int gid  = blockIdx.x * blockDim.x + threadIdx.x;
  int w    = gid >> 5;           // edge index (one wave per edge)
  int lane = threadIdx.x & 31;
  if (w >= E) return;

  int src = esrc[w];
  int dst = edst[w];

  const float4 v = *reinterpret_cast<const float4*>(x_src + (long)src * DIM + lane * 4);
  float* sb = summed + (long)dst * DIM + lane * 4;
  fatomic_add(sb + 0, v.x);
  fatomic_add(sb + 1, v.y);
  fatomic_add(sb + 2, v.z);
  fatomic_add(sb + 3, v.w);
  if (lane == 0) fatomic_add(cnt + dst, 1.0f);
}

// ---------------------------------------------------------------------------
// Kernel 2: fused mean-normalize + dual GEMM + bias using V_WMMA_F32_16X16X4_F32
// out[16 rows x 128 cols] per block; 8 waves, one 16x16 N-tile each.
// Both products share one f32 accumulator (out = agg@Wn + xdst@Ws + bn + bs).
// ---------------------------------------------------------------------------
__global__ __launch_bounds__(256, 2)
void sage_gemm_wmma(const float* __restrict__ summed,
                    const float* __restrict__ cnt,
                    const float* __restrict__ x_dst,
                    const float* __restrict__ Wn,
                    const float* __restrict__ bn,
                    const float* __restrict__ Ws,
                    const float* __restrict__ bs,
                    float*       __restrict__ out,
                    int num_dst) {
  __shared__ __align__(16) float As[2][16][LDST];  // [0]=agg, [1]=x_dst
  __shared__ float invc[16];

  const int tid = threadIdx.x;
  const int m0  = blockIdx.x * 16;

  if (tid < 16) {
    int   g = m0 + tid;
    float c = (g < num_dst) ? cnt[g] : 1.0f;
    invc[tid] = 1.0f / fmaxf(c, 1.0f);
  }
  __syncthreads();

  // Stage A-tiles: 2048 floats per matrix, 8 coalesced elements per thread.
  #pragma unroll
  for (int i = 0; i < 8; ++i) {
    int idx = tid + 256 * i;
    int r   = idx >> 7;        // row 0..15
    int c   = idx & 127;       // col 0..127
    int g   = m0 + r;
    float va = 0.f, vd = 0.f;
    if (g < num_dst) {
      va = summed[(long)g * DIM + c] * invc[r];   // mean applied at load
      vd = x_dst [(long)g * DIM + c];
    }
    As[0][r][c] = va;
    As[1][r][c] = vd;
  }
  __syncthreads();

  const int wave = tid >> 5;
  const int lane = tid & 31;
  const int l15  = lane & 15;
  const int lh   = lane >> 4;        // lane half
  const int n0   = wave * 16;
  const int n    = n0 + l15;

  v8f acc = {};

  #pragma unroll
  for (int kk = 0; kk < DIM / 4; ++kk) {
    // ISA A-layout 16x4 f32: lanes 0-15 hold K={kb,kb+1}, lanes 16-31 K={kb+2,kb+3}
    const int kb = 4 * kk + 2 * lh;  // even -> 8B-aligned LDS float2 reads

    const float2 a0 = *reinterpret_cast<const float2*>(&As[0][l15][kb]);
    const float2 a1 = *reinterpret_cast<const float2*>(&As[1][l15][kb]);
    v2f aN; aN[0] = a0.x; aN[1] = a0.y;
    v2f aS; aS[0] = a1.x; aS[1] = a1.y;

    // B 4x16 f32: row k striped across lanes; 64B-contiguous per lane half.
    v2f bN, bS;
    bN[0] = Wn[(long)(kb    ) * OUTD + n];
    bN[1] = Wn[(long)(kb + 1) * OUTD + n];
    bS[0] = Ws[(long)(kb    ) * OUTD + n];
    bS[1] = Ws[(long)(kb + 1) * OUTD + n];

    acc = __builtin_amdgcn_wmma_f32_16x16x4_f32(false, aN, false, bN,
                                                (short)0, acc, false, false);
    acc = __builtin_amdgcn_wmma_f32_16x16x4_f32(false, aS, false, bS,
                                                (short)0, acc, false, false);
  }

  const float bias = bn[n] + bs[n];
  // C/D layout: vgpr r -> row r (lanes 0-15) / row r+8 (lanes 16-31)
  #pragma unroll
  for (int r = 0; r < 8; ++r) {
    int m = m0 + r + 8 * lh;
    if (m < num_dst) out[(long)m * OUTD + n] = acc[r] + bias;
  }
}

// ---------------------------------------------------------------------------
// Host launcher
// ---------------------------------------------------------------------------
extern "C" void kernel_launch(void* const* d_in, const int* in_sizes, int n_in,
                              void* d_out, int out_size, void* d_ws, size_t ws_size,
                              hipStream_t stream) {
  const float* x_src = (const float*)d_in[0];
  const float* x_dst = (const float*)d_in[1];
  const int*   esrc  = (const int*)  d_in[2];
  const int*   edst  = (const int*)  d_in[3];
  // d_in[4] = num_dst (device scalar) -- derived host-side from in_sizes instead
  const float* Wn    = (const float*)d_in[5];
  const float* bn    = (const float*)d_in[6];
  const float* Ws    = (const float*)d_in[7];
  const float* bs    = (const float*)d_in[8];
  float*       out   = (float*)d_out;

  const int num_dst = in_sizes[1] / DIM;   // x_dst is [num_dst, 128]
  const int E       = in_sizes[2];         // edge_src element count

  float* summed = (float*)d_ws;                       // [num_dst, 128]
  float* cntp   = summed + (size_t)num_dst * DIM;     // [num_dst]

  // 1) zero scatter workspace
  const long nz = (long)num_dst * (DIM + 1);
  zero_ws_kernel<<<2048, 256, 0, stream>>>(summed, nz);

  // 2) edge scatter-sum: one wave per edge
  const long sthreads = (long)E * 32;
  const int  sblocks  = (int)((sthreads + 255) / 256);
  scatter_mean_accum<<<sblocks, 256, 0, stream>>>(x_src, esrc, edst, summed, cntp, E);

  // 3) fused mean + dual GEMM + bias (WMMA f32)
  const int gblocks = (num_dst + 15) / 16;
  sage_gemm_wmma<<<gblocks, 256, 0, stream>>>(summed, cntp, x_dst, Wn, bn, Ws, bs,
                                              out, num_dst);
}